// ANFIS_66666482368771
// MI455X (gfx1250) — compile-verified
//
#include <hip/hip_runtime.h>

#define N_SAMPLES 65536
#define DIM 32
#define RULES 128
#define ROWS_PER_WG 64
#define LDS_STRIDE 132   // 16x128 tile padded (33 float4s per row)

typedef float v2f __attribute__((ext_vector_type(2)));
typedef float v4f __attribute__((ext_vector_type(4)));
typedef float v8f __attribute__((ext_vector_type(8)));

// ---------------------------------------------------------------------------
// Prep: build the quadratic-form coefficient matrices from a, b, c.
//   exponent(n,r) = sum_d w*(x-a)^2 = [x^2, x] . [w ; -2wa] + sum_d w*a^2
//   rule_out(n,r) = [x] . [c_rd] + c_r,32
// Stored k-major (transposed) so WMMA B-fragments are contiguous float2 loads.
// ---------------------------------------------------------------------------
__global__ void anfis_prep(const float* __restrict__ a,
                           const float* __restrict__ b,
                           const float* __restrict__ c,
                           float* __restrict__ Bt1,   // [RULES][64]
                           float* __restrict__ Bt2,   // [RULES][32]
                           float* __restrict__ Kc,    // [RULES]
                           float* __restrict__ bias)  // [RULES]
{
    int r = blockIdx.x * blockDim.x + threadIdx.x;
    if (r >= RULES) return;
    float ksum = 0.0f;
    for (int d = 0; d < DIM; ++d) {
        float bv = fmaxf(b[r * DIM + d], 1e-8f);
        float w  = 1.0f / (2.0f * bv * bv);
        float av = a[r * DIM + d];
        Bt1[r * 64 + d]      = w;               // multiplies x^2  (k = d)
        Bt1[r * 64 + 32 + d] = -2.0f * w * av;  // multiplies x    (k = 32+d)
        Bt2[r * 32 + d]      = c[r * (DIM + 1) + d];
        ksum += w * av * av;
    }
    Kc[r]   = ksum;
    bias[r] = c[r * (DIM + 1) + DIM];
}

// ---------------------------------------------------------------------------
// Main: 256 threads = 8 waves. Wave w owns rule columns [16w, 16w+16).
// Each block processes 64 samples as 4 WMMA row-tiles of 16.
// Outputs are streamed from the LDS tile as coalesced b128 stores.
// ---------------------------------------------------------------------------
__global__ __launch_bounds__(256) void anfis_main(
    const float* __restrict__ X,
    const float* __restrict__ Bt1,
    const float* __restrict__ Bt2,
    const float* __restrict__ Kc,
    const float* __restrict__ bias,
    float* __restrict__ pred,        // [N]
    float* __restrict__ strengths,   // [N][RULES]
    float* __restrict__ normalized)  // [N][RULES]
{
    __shared__ __align__(16) float s_lds[16 * LDS_STRIDE];
    __shared__ __align__(16) float o_lds[16 * LDS_STRIDE];
    __shared__ float p_lds[16 * 16];
    __shared__ float rowsum[16];

    const int tid  = threadIdx.x;
    const int wave = tid >> 5;
    const int lane = tid & 31;
    const int half = (lane < 16) ? 0 : 2;  // K sub-offset per ISA A/B layout
    const int lrow = lane & 15;            // M (for A) / N (for B,C,D)
    const int ncol = wave * 16 + lrow;     // global rule column

    // Preload this wave's B fragments: held in registers across all row tiles.
    v2f b1[16];  // strengths exponent, K = 0..63
    v2f b2[8];   // rule_out,           K = 0..31 (x half only)
#pragma unroll
    for (int ks = 0; ks < 16; ++ks)
        b1[ks] = *(const v2f*)(Bt1 + ncol * 64 + 4 * ks + half);
#pragma unroll
    for (int ks = 0; ks < 8; ++ks)
        b2[ks] = *(const v2f*)(Bt2 + ncol * 32 + 4 * ks + half);
    const float kc = Kc[ncol];
    const float bs = bias[ncol];

    // Epilogue mapping: thread owns row rr, column chunks [4j,4j+4) & [64+4j,+4)
    const int rr = tid >> 4;
    const int j4 = (tid & 15) * 4;

    const int m_base = blockIdx.x * ROWS_PER_WG;

    for (int t = 0; t < 4; ++t) {
        const int m0  = m_base + 16 * t;
        const int row = m0 + lrow;

        // A fragments: lane (0-15) holds K={4ks,4ks+1}, lane (16-31) K={4ks+2,4ks+3}
        v2f xp[8];
#pragma unroll
        for (int ks = 0; ks < 8; ++ks)
            xp[ks] = *(const v2f*)(X + row * DIM + 4 * ks + half);

        if (t < 3)  // warm next tile's X rows
            __builtin_prefetch(X + (row + 16) * DIM, 0, 3);

        v8f acc_s = {0.f, 0.f, 0.f, 0.f, 0.f, 0.f, 0.f, 0.f};
        v8f acc_o = {0.f, 0.f, 0.f, 0.f, 0.f, 0.f, 0.f, 0.f};

        // x^2 half of the exponent GEMM (K = 0..31)
#pragma unroll
        for (int ks = 0; ks < 8; ++ks) {
            v2f a2 = xp[ks] * xp[ks];
            acc_s = __builtin_amdgcn_wmma_f32_16x16x4_f32(
                false, a2, false, b1[ks], (short)0, acc_s, false, false);
        }
        // x half of exponent GEMM (K = 32..63) + full rule_out GEMM
#pragma unroll
        for (int ks = 0; ks < 8; ++ks) {
            acc_s = __builtin_amdgcn_wmma_f32_16x16x4_f32(
                false, xp[ks], false, b1[8 + ks], (short)0, acc_s, false, false);
            acc_o = __builtin_amdgcn_wmma_f32_16x16x4_f32(
                false, xp[ks], false, b2[ks], (short)0, acc_o, false, false);
        }

        __syncthreads();  // previous iteration's LDS readers are done

        // Stash tile in LDS: strengths = exp(-(acc+K)), rule_out = acc + bias.
        const int mrow = (lane < 16) ? 0 : 8;  // C/D layout: VGPR i -> M=i or i+8
#pragma unroll
        for (int i = 0; i < 8; ++i) {
            s_lds[(i + mrow) * LDS_STRIDE + ncol] = __expf(-(acc_s[i] + kc));
            o_lds[(i + mrow) * LDS_STRIDE + ncol] = acc_o[i] + bs;
        }
        __syncthreads();

        // Row-sum partials: two b128 LDS loads per thread.
        v4f s0 = *(const v4f*)&s_lds[rr * LDS_STRIDE + j4];
        v4f s1 = *(const v4f*)&s_lds[rr * LDS_STRIDE + 64 + j4];
        p_lds[rr * 16 + (tid & 15)] =
            (s0[0] + s0[1]) + (s0[2] + s0[3]) + (s1[0] + s1[1]) + (s1[2] + s1[3]);
        __syncthreads();
        if (tid < 16) {
            float rs = 0.0f;
#pragma unroll
            for (int k = 0; k < 16; ++k) rs += p_lds[tid * 16 + k];
            rowsum[tid] = rs + 1e-8f;
        }
        __syncthreads();

        // Outputs: strengths + normalized as coalesced b128 stores; pred partials.
        {
            const float inv = 1.0f / rowsum[rr];
            v4f o0 = *(const v4f*)&o_lds[rr * LDS_STRIDE + j4];
            v4f o1 = *(const v4f*)&o_lds[rr * LDS_STRIDE + 64 + j4];
            v4f n0 = s0 * inv;
            v4f n1 = s1 * inv;

            float* srow = strengths  + (size_t)(m0 + rr) * RULES;
            float* nrow = normalized + (size_t)(m0 + rr) * RULES;
            *(v4f*)&srow[j4]      = s0;
            *(v4f*)&srow[64 + j4] = s1;
            *(v4f*)&nrow[j4]      = n0;
            *(v4f*)&nrow[64 + j4] = n1;

            v4f pv = n0 * o0 + n1 * o1;
            p_lds[rr * 16 + (tid & 15)] = (pv[0] + pv[1]) + (pv[2] + pv[3]);
        }
        __syncthreads();
        if (tid < 16) {
            float pr = 0.0f;
#pragma unroll
            for (int k = 0; k < 16; ++k) pr += p_lds[tid * 16 + k];
            pred[m0 + tid] = pr;
        }
    }
}

extern "C" void kernel_launch(void* const* d_in, const int* in_sizes, int n_in,
                              void* d_out, int out_size, void* d_ws, size_t ws_size,
                              hipStream_t stream) {
    (void)in_sizes; (void)n_in; (void)out_size; (void)ws_size;

    const float* X = (const float*)d_in[0];  // [N, 32]
    const float* a = (const float*)d_in[1];  // [128, 32]
    const float* b = (const float*)d_in[2];  // [128, 32]
    const float* c = (const float*)d_in[3];  // [128, 33]

    float* out        = (float*)d_out;
    float* pred       = out;                              // [N]
    float* strengths  = out + N_SAMPLES;                  // [N,128]
    float* normalized = out + N_SAMPLES + (size_t)N_SAMPLES * RULES;  // [N,128]

    // Workspace layout (floats): Bt1[128*64] | Bt2[128*32] | Kc[128] | bias[128]
    float* ws   = (float*)d_ws;
    float* Bt1  = ws;
    float* Bt2  = ws + RULES * 64;
    float* Kc   = Bt2 + RULES * 32;
    float* bias = Kc + RULES;

    anfis_prep<<<1, 128, 0, stream>>>(a, b, c, Bt1, Bt2, Kc, bias);

    anfis_main<<<N_SAMPLES / ROWS_PER_WG, 256, 0, stream>>>(
        X, Bt1, Bt2, Kc, bias, pred, strengths, normalized);
}